// WhiteMatMul_28406913696455
// MI455X (gfx1250) — compile-verified
//
#include <hip/hip_runtime.h>
#include <stdint.h>

#define TBL_BYTES 65536
#define LDS_BYTES (5 * TBL_BYTES)   // 327680 = 320 KB: mulT + add0..add3 as u8
#define ISPLIT 2                    // grid.y: i-range split for WGP occupancy

// Convert int32 tables (values all < 256) into compact u8 tables in workspace.
// ws layout: [0,64K) mul_table TRANSPOSED (mulT[r*256+l] = mul[l*256+r]),
//            [64K,320K) add_tables[0..3] copied as-is.
__global__ __launch_bounds__(256) void prep_kernel(const int* __restrict__ mul,
                                                   const int* __restrict__ adds,
                                                   uint8_t* __restrict__ ws) {
  int idx = blockIdx.x * 256 + threadIdx.x;
  if (idx < TBL_BYTES) {
    int r = idx >> 8, l = idx & 255;
    ws[idx] = (uint8_t)mul[l * 256 + r];          // transpose
  } else if (idx < LDS_BYTES) {
    ws[idx] = (uint8_t)adds[idx - TBL_BYTES];     // add_tables[0..3]
  }
}

// Block = (b, 32-wide o tile) x i-half, 1024 threads = 32 waves.
// Wave w handles o = ob + w (right column is wave-uniform -> scalar regs).
// Lanes sweep i = i0 + lane + 32*k.
__global__ __launch_bounds__(1024) void lutmm_kernel(
    const int*   __restrict__ left,    // (16,256,64) int32, values < 256
    const int*   __restrict__ right,   // (16,64,256) int32, values < 256
    const int*   __restrict__ adds,    // (5,256,256) int32 (table 4 read from global)
    const float* __restrict__ ftab,    // (256,256) f32
    const uint8_t* __restrict__ ws,    // u8 tables staged by prep_kernel
    float*       __restrict__ out)     // (16,256,256) f32
{
  __shared__ uint8_t smem[LDS_BYTES];

  // ---- stage 320KB of tables into LDS via CDNA5 async global->LDS copies ----
  {
    uint32_t sbase = (uint32_t)(uintptr_t)&smem[0];
    #pragma unroll
    for (int k = 0; k < 20; ++k) {
      uint32_t off = (uint32_t)threadIdx.x * 16u + (uint32_t)k * 16384u;
      uint32_t dst = sbase + off;
      asm volatile("global_load_async_to_lds_b128 %0, %1, %2"
                   :: "v"(dst), "v"(off), "s"(ws)
                   : "memory");
    }
    asm volatile("s_wait_asynccnt 0" ::: "memory");
  }
  __syncthreads();

  const int b    = blockIdx.x >> 3;
  const int ob   = (blockIdx.x & 7) << 5;
  const int i0   = (int)blockIdx.y * (256 / ISPLIT);
  const int lane = (int)(threadIdx.x & 31u);
  const int wv   = __builtin_amdgcn_readfirstlane((int)(threadIdx.x >> 5));
  const int o    = ob + wv;                    // uniform within wave

  // right[b, m, o] for m=0..63, wave-uniform, packed 4 bytes per word.
  uint32_t rqp[16];
  {
    const int* rp = right + (size_t)b * (64 * 256) + o;
    #pragma unroll
    for (int mq = 0; mq < 16; ++mq) {
      uint32_t v0 = (uint32_t)rp[(4 * mq + 0) * 256] & 255u;
      uint32_t v1 = (uint32_t)rp[(4 * mq + 1) * 256] & 255u;
      uint32_t v2 = (uint32_t)rp[(4 * mq + 2) * 256] & 255u;
      uint32_t v3 = (uint32_t)rp[(4 * mq + 3) * 256] & 255u;
      rqp[mq] = v0 | (v1 << 8) | (v2 << 16) | (v3 << 24);
    }
  }

  const int* add4 = adds + 4 * 65536;

  #pragma unroll 1
  for (int k = 0; k < 8 / ISPLIT; ++k) {
    const int i = i0 + lane + (k << 5);
    const int4* lp4 = (const int4*)(left + ((size_t)(b * 256 + i) << 6));
    uint32_t t[4];
    // The reduction tree is pairwise-adjacent, so each aligned 16-wide m-chunk
    // reduces independently to one byte through levels 0..3.
    #pragma unroll
    for (int c = 0; c < 4; ++c) {
      int4 A = lp4[4 * c + 0];
      int4 B = lp4[4 * c + 1];
      int4 C = lp4[4 * c + 2];
      int4 D = lp4[4 * c + 3];
      uint32_t lv[16] = { (uint32_t)A.x, (uint32_t)A.y, (uint32_t)A.z, (uint32_t)A.w,
                          (uint32_t)B.x, (uint32_t)B.y, (uint32_t)B.z, (uint32_t)B.w,
                          (uint32_t)C.x, (uint32_t)C.y, (uint32_t)C.z, (uint32_t)C.w,
                          (uint32_t)D.x, (uint32_t)D.y, (uint32_t)D.z, (uint32_t)D.w };
      uint32_t p[16];
      #pragma unroll
      for (int j = 0; j < 16; ++j) {
        uint32_t r = (rqp[4 * c + (j >> 2)] >> ((j & 3) * 8)) & 255u;
        // mulT[r][l]: r uniform, l random per lane -> banks spread by l>>2
        p[j] = smem[(r << 8) | lv[j]];
      }
      uint32_t q0[8];
      #pragma unroll
      for (int j = 0; j < 8; ++j)
        q0[j] = smem[1 * TBL_BYTES + ((p[2 * j] << 8) | p[2 * j + 1])];
      uint32_t q1[4];
      #pragma unroll
      for (int j = 0; j < 4; ++j)
        q1[j] = smem[2 * TBL_BYTES + ((q0[2 * j] << 8) | q0[2 * j + 1])];
      uint32_t q2a = smem[3 * TBL_BYTES + ((q1[0] << 8) | q1[1])];
      uint32_t q2b = smem[3 * TBL_BYTES + ((q1[2] << 8) | q1[3])];
      t[c] = smem[4 * TBL_BYTES + ((q2a << 8) | q2b)];
    }
    // level 4 + final from global (L2-resident)
    uint32_t a  = (uint32_t)add4[(t[0] << 8) | t[1]] & 255u;
    uint32_t b2 = (uint32_t)add4[(t[2] << 8) | t[3]] & 255u;
    out[((size_t)(b * 256 + i) << 8) + (size_t)o] = ftab[(a << 8) | b2];
  }
}

extern "C" void kernel_launch(void* const* d_in, const int* in_sizes, int n_in,
                              void* d_out, int out_size, void* d_ws, size_t ws_size,
                              hipStream_t stream) {
  const int*   left  = (const int*)d_in[0];
  const int*   right = (const int*)d_in[1];
  const int*   mul   = (const int*)d_in[2];
  const int*   adds  = (const int*)d_in[3];
  const float* ftab  = (const float*)d_in[4];
  float*   out = (float*)d_out;
  uint8_t* ws  = (uint8_t*)d_ws;

  (void)in_sizes; (void)n_in; (void)out_size; (void)ws_size;

  // Convert/transpose tables to u8 in workspace (320 KB), then main kernel.
  prep_kernel<<<LDS_BYTES / 256, 256, 0, stream>>>(mul, adds, ws);
  dim3 grid(128, ISPLIT);
  lutmm_kernel<<<grid, 1024, 0, stream>>>(left, right, adds, ftab, ws, out);
}